// GAT_20916490731923
// MI455X (gfx1250) — compile-verified
//
#include <hip/hip_runtime.h>
#include <math.h>

// ---------------------------------------------------------------------------
// GAT 2-layer inference for MI455X (gfx1250, wave32).
//  N=100000 nodes, IN=128, layer1: 8 heads x 8 ch (concat->64), layer2: 1x32.
//  GEMMs use V_WMMA_F32_16X16X4_F32 (fp32 matrix pipe).
//  Edge softmax/aggregation via L2-resident fp32 global atomics
//  (91MB working set << 192MB L2).
// ---------------------------------------------------------------------------

typedef float v2f __attribute__((ext_vector_type(2)));
typedef float v8f __attribute__((ext_vector_type(8)));

#define LL long long

// ---------------- generic fill ----------------
__global__ void fill_kernel(float* __restrict__ p, LL n, float v) {
    LL t = (LL)blockIdx.x * blockDim.x + threadIdx.x;
    if (t < n) p[t] = v;
}

// ---------------- fp32 WMMA GEMM:  H[M,N] = X[M,K] @ W[K,N] ----------------
// One wave computes one 16x16 output tile via V_WMMA_F32_16X16X4_F32,
// stepping K by 4. M must be a multiple of 16 (100000 = 6250*16).
// VGPR layouts per CDNA5 ISA §7.12.2:
//   A 16x4 : lanes 0-15 -> {K=k0,k0+1}, lanes 16-31 -> {K=k0+2,k0+3}, row = lane&15
//   B 4x16 : VGPR0 -> K=k0(+2 for hi lanes), VGPR1 -> K=k0+1(+3), col = lane&15
//   C/D    : VGPR v -> row v + (lane<16 ? 0 : 8), col = lane&15
__global__ void gemm_wmma_f32(const float* __restrict__ X,
                              const float* __restrict__ W,
                              float* __restrict__ H,
                              int M, int N, int K) {
    const int wave = (int)((blockIdx.x * (LL)blockDim.x + threadIdx.x) >> 5);
    const int lane = threadIdx.x & 31;
    const int ntiles = N >> 4;
    const int totalWaves = (M >> 4) * ntiles;
    if (wave >= totalWaves) return;            // wave-uniform: EXEC stays full

    const int tm = wave / ntiles;
    const int tn = wave % ntiles;
    const int row   = tm * 16 + (lane & 15);   // A row this lane supplies
    const int col   = tn * 16 + (lane & 15);   // B/C/D column this lane owns
    const int khalf = (lane >> 4) * 2;         // 0 for lanes 0-15, 2 for 16-31

    v8f acc = {};
    for (int k0 = 0; k0 < K; k0 += 4) {
        v2f a, b;
        a.x = X[row * K + k0 + khalf + 0];
        a.y = X[row * K + k0 + khalf + 1];
        b.x = W[(k0 + khalf + 0) * N + col];
        b.y = W[(k0 + khalf + 1) * N + col];
        acc = __builtin_amdgcn_wmma_f32_16x16x4_f32(
            /*neg_a=*/false, a, /*neg_b=*/false, b,
            /*c_mod=*/(short)0, acc, /*reuse_a=*/false, /*reuse_b=*/false);
    }

    const int rbase = tm * 16 + ((lane >> 4) << 3);   // 0 or +8 row group
#pragma unroll
    for (int v = 0; v < 8; ++v)
        H[(LL)(rbase + v) * N + col] = acc[v];
}

// ---------------- per-(node,head) attention coefficients ----------------
// al_s[n,h] = sum_c feat[n, h*C + c] * a_src[h,c]   (same for a_dst)
__global__ void attn_kernel(const float* __restrict__ feat,
                            const float* __restrict__ a_src,
                            const float* __restrict__ a_dst,
                            float* __restrict__ al_s, float* __restrict__ al_d,
                            LL N, int Hh, int C) {
    LL t = (LL)blockIdx.x * blockDim.x + threadIdx.x;
    if (t >= N * Hh) return;
    int h = (int)(t % Hh);
    const float* f = feat + t * C;      // (n*Hh + h)*C
    float ss = 0.f, sd = 0.f;
    for (int c = 0; c < C; ++c) {
        float fv = f[c];
        ss += fv * a_src[h * C + c];
        sd += fv * a_dst[h * C + c];
    }
    al_s[t] = ss;
    al_d[t] = sd;
}

// ---------------- float atomic max (sign-magnitude trick) ----------------
__device__ __forceinline__ void atomicMaxFloat(float* addr, float val) {
    if (val >= 0.f) atomicMax((int*)addr, __float_as_int(val));
    else            atomicMin((unsigned int*)addr, __float_as_uint(val));
}

__device__ __forceinline__ void edge_ids(const LL* __restrict__ ei, LL E, LL e,
                                         LL& src, LL& dst) {
    if (e < E) { src = ei[e]; dst = ei[E + e]; }
    else       { src = dst = e - E; }          // appended self-loop
}

// ---------------- edge pass 1: segment max of leaky-relu logits ----------
__global__ void edge_max_kernel(const LL* __restrict__ ei, LL E, LL N, int Hh,
                                const float* __restrict__ al_s,
                                const float* __restrict__ al_d,
                                float* __restrict__ m) {
    LL t = (LL)blockIdx.x * blockDim.x + threadIdx.x;
    LL total = (E + N) * Hh;
    if (t >= total) return;
    LL e = t / Hh; int h = (int)(t % Hh);
    LL src, dst; edge_ids(ei, E, e, src, dst);
    float v = al_s[src * Hh + h] + al_d[dst * Hh + h];
    v = v > 0.f ? v : 0.2f * v;                // leaky relu, slope 0.2
    atomicMaxFloat(&m[dst * Hh + h], v);
}

// ---------------- edge pass 2: p=exp(e-m); s[dst]+=p; agg[dst]+=h[src]*p --
__global__ void edge_accum_kernel(const LL* __restrict__ ei, LL E, LL N,
                                  int Hh, int C,
                                  const float* __restrict__ feat,
                                  const float* __restrict__ al_s,
                                  const float* __restrict__ al_d,
                                  const float* __restrict__ m,
                                  float* __restrict__ s,
                                  float* __restrict__ agg) {
    LL t = (LL)blockIdx.x * blockDim.x + threadIdx.x;
    LL total = (E + N) * Hh;
    if (t >= total) return;
    LL e = t / Hh; int h = (int)(t % Hh);
    LL src, dst; edge_ids(ei, E, e, src, dst);
    float v = al_s[src * Hh + h] + al_d[dst * Hh + h];
    v = v > 0.f ? v : 0.2f * v;
    float p = expf(v - m[dst * Hh + h]);
    atomicAdd(&s[dst * Hh + h], p);
    const float* fs = feat + (src * Hh + h) * C;
    float*       ad = agg  + (dst * Hh + h) * C;
    for (int c = 0; c < C; ++c) atomicAdd(&ad[c], fs[c] * p);
}

// ---------------- layer-1 finalize: /(s+eps) + bias, ELU, in place --------
__global__ void finalize1_kernel(const float* __restrict__ agg,
                                 const float* __restrict__ s,
                                 const float* __restrict__ b,
                                 float* __restrict__ out,
                                 LL N, int Hh, int C) {
    LL t = (LL)blockIdx.x * blockDim.x + threadIdx.x;
    if (t >= N * Hh) return;
    int h = (int)(t % Hh);
    float inv = 1.f / (s[t] + 1e-16f);
    for (int c = 0; c < C; ++c) {
        float v = agg[t * C + c] * inv + b[h * C + c];
        out[t * C + c] = v > 0.f ? v : (expf(v) - 1.f);   // ELU
    }
}

// ---------------- layer-2 finalize: bias + log_softmax over 32 classes ----
__global__ void finalize2_kernel(const float* __restrict__ agg,
                                 const float* __restrict__ s,
                                 const float* __restrict__ b,
                                 float* __restrict__ out, LL N, int C) {
    LL n = (LL)blockIdx.x * blockDim.x + threadIdx.x;
    if (n >= N) return;
    float inv = 1.f / (s[n] + 1e-16f);
    float logits[32];
    float mx = -INFINITY;
    for (int c = 0; c < C; ++c) {
        float v = agg[n * C + c] * inv + b[c];
        logits[c] = v;
        mx = v > mx ? v : mx;
    }
    float sum = 0.f;
    for (int c = 0; c < C; ++c) sum += expf(logits[c] - mx);
    float lse = mx + logf(sum);
    for (int c = 0; c < C; ++c) out[n * C + c] = logits[c] - lse;
}

// ---------------------------------------------------------------------------
extern "C" void kernel_launch(void* const* d_in, const int* in_sizes, int n_in,
                              void* d_out, int out_size, void* d_ws, size_t ws_size,
                              hipStream_t stream) {
    const float* x    = (const float*)d_in[0];
    const LL*    ei   = (const LL*)   d_in[1];   // int64 [2,E]
    const float* W1   = (const float*)d_in[2];
    const float* a_s1 = (const float*)d_in[3];
    const float* a_d1 = (const float*)d_in[4];
    const float* b1   = (const float*)d_in[5];
    const float* W2   = (const float*)d_in[6];
    const float* a_s2 = (const float*)d_in[7];
    const float* a_d2 = (const float*)d_in[8];
    const float* b2   = (const float*)d_in[9];
    float* out = (float*)d_out;

    const LL N = in_sizes[0] / 128;     // 100000
    const LL E = in_sizes[1] / 2;       // 1600000
    const int IN = 128, H1 = 8, C1 = 8, F1 = 64, C2 = 32;

    // ---- workspace layout (fp32) ----
    float* ws   = (float*)d_ws;
    float* h1   = ws;                       // N*64 (pre-attn feats, then post-ELU)
    float* als1 = h1   + (size_t)N * F1;    // N*8
    float* ald1 = als1 + (size_t)N * H1;    // N*8
    float* m1   = ald1 + (size_t)N * H1;    // N*8
    float* s1   = m1   + (size_t)N * H1;    // N*8
    float* agg1 = s1   + (size_t)N * H1;    // N*64
    float* h2   = agg1 + (size_t)N * F1;    // N*32
    float* als2 = h2   + (size_t)N * C2;    // N
    float* ald2 = als2 + (size_t)N;         // N
    float* m2   = ald2 + (size_t)N;         // N
    float* s2   = m2   + (size_t)N;         // N
    float* agg2 = s2   + (size_t)N;         // N*32

    const int BLK = 256;
    auto blocks = [](LL n, int blk) { return (unsigned)((n + blk - 1) / blk); };

    // ===================== layer 1 =====================
    fill_kernel<<<blocks(N * H1, BLK), BLK, 0, stream>>>(m1, N * H1, -INFINITY);
    // s1 and agg1 are contiguous: zero both in one launch
    fill_kernel<<<blocks(N * (H1 + F1), BLK), BLK, 0, stream>>>(s1, N * (H1 + F1), 0.f);

    // h1 = x @ W1  (WMMA fp32): 6250 * 4 tiles, 1 wave each, 4 waves/block
    {
        LL waves = (N / 16) * (F1 / 16);
        gemm_wmma_f32<<<blocks(waves * 32, 128), 128, 0, stream>>>(
            x, W1, h1, (int)N, F1, IN);
    }
    attn_kernel<<<blocks(N * H1, BLK), BLK, 0, stream>>>(
        h1, a_s1, a_d1, als1, ald1, N, H1, C1);
    edge_max_kernel<<<blocks((E + N) * H1, BLK), BLK, 0, stream>>>(
        ei, E, N, H1, als1, ald1, m1);
    edge_accum_kernel<<<blocks((E + N) * H1, BLK), BLK, 0, stream>>>(
        ei, E, N, H1, C1, h1, als1, ald1, m1, s1, agg1);
    finalize1_kernel<<<blocks(N * H1, BLK), BLK, 0, stream>>>(
        agg1, s1, b1, h1, N, H1, C1);          // ELU(result) written over h1

    // ===================== layer 2 =====================
    fill_kernel<<<blocks(N, BLK), BLK, 0, stream>>>(m2, N, -INFINITY);
    // s2 and agg2 are contiguous: zero both in one launch
    fill_kernel<<<blocks(N * (1 + C2), BLK), BLK, 0, stream>>>(s2, N * (1 + C2), 0.f);

    // h2 = h1' @ W2  (WMMA fp32): 6250 * 2 tiles
    {
        LL waves = (N / 16) * (C2 / 16);
        gemm_wmma_f32<<<blocks(waves * 32, 128), 128, 0, stream>>>(
            h1, W2, h2, (int)N, C2, F1);
    }
    attn_kernel<<<blocks(N, BLK), BLK, 0, stream>>>(
        h2, a_s2, a_d2, als2, ald2, N, 1, C2);
    edge_max_kernel<<<blocks(E + N, BLK), BLK, 0, stream>>>(
        ei, E, N, 1, als2, ald2, m2);
    edge_accum_kernel<<<blocks(E + N, BLK), BLK, 0, stream>>>(
        ei, E, N, 1, C2, h2, als2, ald2, m2, s2, agg2);
    finalize2_kernel<<<blocks(N, BLK), BLK, 0, stream>>>(
        agg2, s2, b2, out, N, C2);
}